// ChamferLoss_53094385713845
// MI455X (gfx1250) — compile-verified
//
#include <hip/hip_runtime.h>
#include <cstddef>

typedef __attribute__((ext_vector_type(2))) float v2f;
typedef __attribute__((ext_vector_type(8))) float v8f;

#define BATCH       4
#define NPTS        8192
#define WAVES_PER_WG 8
#define ROWS_PER_WG (WAVES_PER_WG * 16)   // 128 rows per workgroup
#define CHUNK       2048                  // columns staged in LDS per pass (24 KB)

// Directional Chamfer term: for each "row" point, min over all "col" points of
// squared distance; sum of mins atomically added into *out.
//   P[i,j] = xx_i + yy_j - 2*g_i.p_j
//   sum_i min_j P = sum_i ( xx_i + min_j ( yy_j - 2*g_i.p_j ) )
// The K=4 padding slot carries the rank-1 yy broadcast:
//   A = (-2gx, -2gy, -2gz, 1),  B = (px, py, pz, yy_j),  C = 0
//   ==>  acc[i][j] = yy_j - 2*g_i.p_j   with no per-tile C-splat movs.
__global__ __launch_bounds__(256) void chamfer_dir_53094385713845(
    const float* __restrict__ rows_pts,   // [BATCH, NPTS, 3]
    const float* __restrict__ cols_pts,   // [BATCH, NPTS, 3]
    float* __restrict__ out)
{
    __shared__ float sB[CHUNK * 3];

    const int tid  = threadIdx.x;
    const int lane = tid & 31;
    const int wave = tid >> 5;
    const int m    = lane & 15;   // row (for A) / column (for B,C,D) within tile
    const int hh   = lane >> 4;   // half-wave: selects K=0,1 vs K=2,3 fragments

    const int bpb = NPTS / ROWS_PER_WG;       // blocks per batch = 64
    const int b   = blockIdx.x / bpb;
    const int rb  = blockIdx.x % bpb;
    const int i0  = rb * ROWS_PER_WG + wave * 16;

    // ---- A fragment: 16 rows of this wave; K slots (-2gx,-2gy,-2gz,1) ----
    const float* rp = rows_pts + ((size_t)b * NPTS + (size_t)(i0 + m)) * 3;
    const float gx = rp[0], gy = rp[1], gz = rp[2];
    const float xx = fmaf(gx, gx, fmaf(gy, gy, gz * gz));
    v2f afrag;
    afrag.x = hh ? (-2.0f * gz) : (-2.0f * gx);  // K=2 : K=0
    afrag.y = hh ? 1.0f         : (-2.0f * gy);  // K=3 : K=1

    // Running min over columns j of (yy_j - 2 g.p), per accumulator row slot.
    float rm[8];
#pragma unroll
    for (int r = 0; r < 8; ++r) rm[r] = __builtin_inff();

    const v8f czero = {};  // loop-invariant zero accumulator input

    const float4* csrc = (const float4*)(cols_pts + (size_t)b * NPTS * 3);

    for (int c0 = 0; c0 < NPTS; c0 += CHUNK) {
        __syncthreads();
        // Cooperative stage of CHUNK points (CHUNK*3 floats) into LDS, 128b ops.
#pragma unroll
        for (int t = 0; t < (CHUNK * 3) / 4; t += 256) {
            ((float4*)sB)[t + tid] = csrc[(size_t)(c0 * 3) / 4 + t + tid];
        }
        __syncthreads();

        const float* pbase = &sB[m * 3];   // hoist lane's column offset
#pragma unroll 4
        for (int jt = 0; jt < CHUNK; jt += 16) {
            const float* pp = pbase + jt * 3;
            const float px = pp[0], py = pp[1], pz = pp[2];
            const float yy = fmaf(px, px, fmaf(py, py, pz * pz));

            v2f bfrag;
            bfrag.x = hh ? pz : px;   // K=2 : K=0
            bfrag.y = hh ? yy : py;   // K=3 : K=1  (yy rides the padding slot)

            v8f acc = __builtin_amdgcn_wmma_f32_16x16x4_f32(
                /*neg_a=*/false, afrag, /*neg_b=*/false, bfrag,
                /*c_mod=*/(short)0, czero, /*reuse_a=*/false, /*reuse_b=*/false);

#pragma unroll
            for (int r = 0; r < 8; ++r) rm[r] = fminf(rm[r], acc[r]);
        }
    }

    // Row-min across columns: D layout puts column j = lane&15; rows r (+8 for
    // upper half). Min-reduce across the 16 lanes of each half (masks 1,2,4,8).
#pragma unroll
    for (int r = 0; r < 8; ++r) {
        float v = rm[r];
        v = fminf(v, __shfl_xor(v, 1, 32));
        v = fminf(v, __shfl_xor(v, 2, 32));
        v = fminf(v, __shfl_xor(v, 4, 32));
        v = fminf(v, __shfl_xor(v, 8, 32));
        rm[r] = v;
    }

    // Add xx_i per row and sum. Lane L holds xx for row (L&15); row r+8*hh
    // lives at source lane r + 8*hh. All lanes run the shuffles (full EXEC),
    // lanes 0 and 16 commit the two half-results.
    float partial = 0.0f;
#pragma unroll
    for (int r = 0; r < 8; ++r) {
        const float xr = __shfl(xx, r + 8 * hh, 32);
        partial += rm[r] + xr;
    }
    if (m == 0) {
        unsafeAtomicAdd(out, partial);   // global_atomic_add_f32
    }
}

extern "C" void kernel_launch(void* const* d_in, const int* in_sizes, int n_in,
                              void* d_out, int out_size, void* d_ws, size_t ws_size,
                              hipStream_t stream) {
    (void)in_sizes; (void)n_in; (void)out_size; (void)d_ws; (void)ws_size;
    const float* gts   = (const float*)d_in[0];   // [4, 8192, 3] f32
    const float* preds = (const float*)d_in[1];   // [4, 8192, 3] f32
    float* out = (float*)d_out;                   // scalar f32

    hipMemsetAsync(out, 0, sizeof(float), stream);

    const dim3 block(256);
    const dim3 grid(BATCH * (NPTS / ROWS_PER_WG));  // 4 * 64 = 256 WGs

    // loss_2: for each gts point, min over preds
    chamfer_dir_53094385713845<<<grid, block, 0, stream>>>(gts, preds, out);
    // loss_1: for each preds point, min over gts
    chamfer_dir_53094385713845<<<grid, block, 0, stream>>>(preds, gts, out);
}